// MXFPQuantizer_42734924595710
// MI455X (gfx1250) — compile-verified
//
#include <hip/hip_runtime.h>

typedef float f32x4 __attribute__((ext_vector_type(4)));

// E2M1 quantize-dequantize of one value given power-of-two scale (s) and its
// reciprocal (inv_s). RNE to 1 mantissa bit via integer bit trick, clamp to
// [0.5, 6] with v_med3, flush |xd| <= 0.25 to zero, restore sign, rescale.
__device__ __forceinline__ float mxfp4_quant1(float x, float inv_s, float s) {
  float xd = x * inv_s;
  unsigned b = __float_as_uint(xd);
  unsigned a = b & 0x7fffffffu;
  float af = __uint_as_float(a);
  unsigned r = a + (((a >> 22) & 1u) + 0x1FFFFFu); // RNE round into bit 22
  r &= 0xFFC00000u;                                // keep sign+exp+1 man bit
  float q = __builtin_amdgcn_fmed3f(__uint_as_float(r), 0.5f, 6.0f);
  q = (af <= 0.25f) ? 0.0f : q;
  float o = q * s;
  return __uint_as_float(__float_as_uint(o) | (b & 0x80000000u));
}

// scale = 2^(floor(log2(amax)) - 2), built straight from exponent bits.
__device__ __forceinline__ void scale_from_amax(float am, float& inv_s, float& s) {
  int e = (int)(__float_as_uint(am) >> 23) - 127;
  int pot = e - 2;
  pot = pot < -126 ? -126 : (pot > 126 ? 126 : pot);
  inv_s = __uint_as_float((unsigned)(127 - pot) << 23);
  s     = __uint_as_float((unsigned)(127 + pot) << 23);
}

// Main streaming kernel. Chunk = 512 floats per wave (4 units of 32 lanes x
// float4). Input staged via async global->LDS (ASYNCcnt double buffer, wave-
// private slices, no barriers); group-of-32 amax via ds_swizzle xor butterfly
// over 8 lanes; NT stores for the output stream.
__global__ __launch_bounds__(256) void mxfp4_kernel(const float* __restrict__ xp,
                                                    float* __restrict__ op,
                                                    unsigned chunksTotal) {
  constexpr unsigned UNITS = 4; // float4-units per chunk
  constexpr unsigned CPW   = 4; // chunks per wave
  __shared__ f32x4 tile[2][8][UNITS][32]; // 32 KB: [buf][wave][unit][lane]

  const unsigned lane = threadIdx.x & 31u;
  const unsigned wave = threadIdx.x >> 5;
  const unsigned c0   = (blockIdx.x * 8u + wave) * CPW;

  unsigned ldsB[2];
  ldsB[0] = (unsigned)(uintptr_t)(&tile[0][wave][0][lane]);
  ldsB[1] = (unsigned)(uintptr_t)(&tile[1][wave][0][lane]);

  auto issue = [&](unsigned c, unsigned ldsBase) {
    unsigned goff = c * 2048u + lane * 16u; // byte offset of this lane's data
#pragma unroll
    for (unsigned u = 0; u < UNITS; ++u) {
      asm volatile("global_load_async_to_lds_b128 %0, %1, %2 th:TH_LOAD_NT"
                   :: "v"(ldsBase + u * 512u), "v"(goff + u * 512u), "s"(xp)
                   : "memory");
    }
  };

  if (c0 < chunksTotal) issue(c0, ldsB[0]);

#pragma unroll 1
  for (unsigned i = 0; i < CPW; ++i) {
    unsigned c = c0 + i;
    if (c >= chunksTotal) break;
    if ((i + 1u < CPW) && (c + 1u < chunksTotal)) {
      issue(c + 1u, ldsB[(i + 1u) & 1u]); // prefetch next chunk into other buf
      asm volatile("s_wait_asynccnt 4" ::: "memory"); // current chunk landed
    } else {
      asm volatile("s_wait_asynccnt 0" ::: "memory");
    }
    const unsigned buf = i & 1u;
#pragma unroll
    for (unsigned u = 0; u < UNITS; ++u) {
      f32x4 v = tile[buf][wave][u][lane];
      float am = fmaxf(fmaxf(__builtin_fabsf(v.x), __builtin_fabsf(v.y)),
                       fmaxf(__builtin_fabsf(v.z), __builtin_fabsf(v.w)));
      // group of 32 floats == 8 consecutive lanes: xor-butterfly max
      am = fmaxf(am, __int_as_float(__builtin_amdgcn_ds_swizzle(__float_as_int(am), 0x041F)));
      am = fmaxf(am, __int_as_float(__builtin_amdgcn_ds_swizzle(__float_as_int(am), 0x081F)));
      am = fmaxf(am, __int_as_float(__builtin_amdgcn_ds_swizzle(__float_as_int(am), 0x101F)));
      float inv_s, s;
      scale_from_amax(am, inv_s, s);
      f32x4 r;
      r.x = mxfp4_quant1(v.x, inv_s, s);
      r.y = mxfp4_quant1(v.y, inv_s, s);
      r.z = mxfp4_quant1(v.z, inv_s, s);
      r.w = mxfp4_quant1(v.w, inv_s, s);
      unsigned idx = c * 512u + u * 128u + lane * 4u;
      __builtin_nontemporal_store(r, (f32x4*)(op + idx)); // th:TH_STORE_NT
    }
  }
}

// Scalar tail for n % 512 != 0 (unused for the reference shape; kept for
// generality). One thread per group of <=32 elements.
__global__ void mxfp4_tail(const float* __restrict__ xp, float* __restrict__ op,
                           unsigned start, unsigned n) {
  unsigned g = blockIdx.x * blockDim.x + threadIdx.x;
  unsigned long long base = (unsigned long long)start + (unsigned long long)g * 32ull;
  if (base >= n) return;
  unsigned cnt = (unsigned)(n - base);
  cnt = cnt > 32u ? 32u : cnt;
  float am = 0.f;
  for (unsigned i = 0; i < cnt; ++i) am = fmaxf(am, __builtin_fabsf(xp[base + i]));
  float inv_s, s;
  scale_from_amax(am, inv_s, s);
  for (unsigned i = 0; i < cnt; ++i) op[base + i] = mxfp4_quant1(xp[base + i], inv_s, s);
}

extern "C" void kernel_launch(void* const* d_in, const int* in_sizes, int n_in,
                              void* d_out, int out_size, void* d_ws, size_t ws_size,
                              hipStream_t stream) {
  (void)n_in; (void)out_size; (void)d_ws; (void)ws_size;
  const float* x = (const float*)d_in[0];
  float* out = (float*)d_out;
  unsigned n = (unsigned)in_sizes[0];

  unsigned chunks = n / 512u; // 512 floats per wave-chunk
  if (chunks > 0u) {
    unsigned waves  = (chunks + 3u) / 4u; // CPW = 4
    unsigned blocks = (waves + 7u) / 8u;  // 8 waves per block
    mxfp4_kernel<<<dim3(blocks), dim3(256), 0, stream>>>(x, out, chunks);
  }
  unsigned start = chunks * 512u;
  if (start < n) {
    unsigned groups = (n - start + 31u) / 32u;
    unsigned blocks = (groups + 255u) / 256u;
    mxfp4_tail<<<dim3(blocks), dim3(256), 0, stream>>>(x, out, start, n);
  }
}